// AutoRegressiveDecoderLayer_7009386627329
// MI455X (gfx1250) — compile-verified
//
#include <hip/hip_runtime.h>
#include <hip/hip_bf16.h>
#include <math.h>

#define DIM   128
#define HEADS 8
#define DH    16
#define BSZ   1024
#define NODES 1000
#define TPREV 511

typedef __attribute__((ext_vector_type(2))) float v2f;
typedef __attribute__((ext_vector_type(8))) float v8f;

__device__ __forceinline__ float wred_sum(float v) {
#pragma unroll
  for (int off = 16; off >= 1; off >>= 1) v += __shfl_xor(v, off, 32);
  return v;
}
__device__ __forceinline__ float wred_max(float v) {
#pragma unroll
  for (int off = 16; off >= 1; off >>= 1) v = fmaxf(v, __shfl_xor(v, off, 32));
  return v;
}

// out[M=1024,128] = act(A[1024,128] @ W[128,128] + bias (+ R))
// One wave per 16x16 tile, f32 WMMA, K=128 in 32 steps of 4.
template <int RELU, int HASRES>
__global__ __launch_bounds__(256) void gemm128_wmma(
    const float* __restrict__ A, const float* __restrict__ W,
    const float* __restrict__ bias, const float* __restrict__ R,
    float* __restrict__ out) {
  const int wave = threadIdx.x >> 5;
  const int lane = threadIdx.x & 31;
  const int row0 = blockIdx.x << 4;
  const int col0 = wave << 4;
  const int r    = lane & 15;
  const int hi   = lane >> 4;        // 0: K=0,1 half / 1: K=2,3 half
  const int koff = hi << 1;
  const float* ap = A + (size_t)(row0 + r) * DIM + koff;
  const float* wp = W + (size_t)koff * DIM + col0 + r;
  v8f acc = {0.f, 0.f, 0.f, 0.f, 0.f, 0.f, 0.f, 0.f};
#pragma unroll
  for (int k = 0; k < DIM; k += 4) {
    v2f a = *(const v2f*)(ap + k);          // A[row, k+koff .. +1]
    v2f b;
    b.x = wp[(size_t)k * DIM];              // W[k+koff,   col]
    b.y = wp[(size_t)k * DIM + DIM];        // W[k+koff+1, col]
    acc = __builtin_amdgcn_wmma_f32_16x16x4_f32(false, a, false, b, (short)0,
                                                acc, false, false);
  }
  const int col  = col0 + r;
  const int rowb = row0 + (hi << 3);
  const float bc = bias[col];
#pragma unroll
  for (int i = 0; i < 8; ++i) {
    float v = acc[i] + bc;
    if (HASRES) v += R[(size_t)(rowb + i) * DIM + col];
    if (RELU) v = fmaxf(v, 0.f);
    out[(size_t)(rowb + i) * DIM + col] = v;
  }
}

// Single-query self-attention over 511 cached tokens + 1 appended token.
// One block per batch row; wave = head; lane strides tokens; online softmax.
__global__ __launch_bounds__(256) void self_attn_kernel(
    const float* __restrict__ q, const float* __restrict__ Kprev,
    const float* __restrict__ Vprev, const float* __restrict__ knew,
    const float* __restrict__ vnew, float* __restrict__ out) {
  __shared__ float red[HEADS][32][DH];
  const int b = blockIdx.x;
  const int wave = threadIdx.x >> 5;
  const int lane = threadIdx.x & 31;
  const int c0 = wave * DH;
  const float* qp = q + (size_t)b * DIM + c0;
  float qv[DH];
#pragma unroll
  for (int j = 0; j < DH; ++j) qv[j] = qp[j];
  const float* Kb = Kprev + (size_t)b * TPREV * DIM + c0;
  const float* Vb = Vprev + (size_t)b * TPREV * DIM + c0;
  const float* kn = knew + (size_t)b * DIM + c0;
  const float* vn = vnew + (size_t)b * DIM + c0;
  float m = -1e30f, l = 0.f, acc[DH];
#pragma unroll
  for (int j = 0; j < DH; ++j) acc[j] = 0.f;
  for (int i = 0; i < 16; ++i) {
    const int t = lane + (i << 5);          // 0..511
    const float* kr = (t < TPREV) ? (Kb + (size_t)t * DIM) : kn;
    const float* vr = (t < TPREV) ? (Vb + (size_t)t * DIM) : vn;
    if (t + 32 < TPREV) {
      __builtin_prefetch(Kb + (size_t)(t + 32) * DIM, 0, 0);
      __builtin_prefetch(Vb + (size_t)(t + 32) * DIM, 0, 0);
    }
    float kk[DH];
#pragma unroll
    for (int j = 0; j < 4; ++j) {
      float4 tv = ((const float4*)kr)[j];
      kk[4 * j] = tv.x; kk[4 * j + 1] = tv.y; kk[4 * j + 2] = tv.z; kk[4 * j + 3] = tv.w;
    }
    float s = 0.f;
#pragma unroll
    for (int j = 0; j < DH; ++j) s = fmaf(qv[j], kk[j], s);
    s *= 0.25f;                              // 1/sqrt(16)
    const float mn = fmaxf(m, s);
    const float p  = __expf(s - mn);
    const float co = __expf(m - mn);
    float vv[DH];
#pragma unroll
    for (int j = 0; j < 4; ++j) {
      float4 tv = ((const float4*)vr)[j];
      vv[4 * j] = tv.x; vv[4 * j + 1] = tv.y; vv[4 * j + 2] = tv.z; vv[4 * j + 3] = tv.w;
    }
    l = l * co + p;
#pragma unroll
    for (int j = 0; j < DH; ++j) acc[j] = fmaf(acc[j], co, p * vv[j]);
    m = mn;
  }
  const float M = wred_max(m);
  const float f = __expf(m - M);
  const float L = wred_sum(l * f);
#pragma unroll
  for (int j = 0; j < DH; ++j) red[wave][lane][j] = acc[j] * f;
  __syncthreads();
  if (lane < DH) {
    float o = 0.f;
#pragma unroll 8
    for (int i = 0; i < 32; ++i) o += red[wave][i][lane];
    out[(size_t)b * DIM + c0 + lane] = o / L;
  }
}

// Single-query cross-attention over 1000 nodes with visited mask.
__global__ __launch_bounds__(256) void cross_attn_kernel(
    const float* __restrict__ q, const float* __restrict__ Katt,
    const float* __restrict__ Vatt, const int* __restrict__ mask,
    float* __restrict__ out) {
  __shared__ float red[HEADS][32][DH];
  const int b = blockIdx.x;
  const int wave = threadIdx.x >> 5;
  const int lane = threadIdx.x & 31;
  const int c0 = wave * DH;
  const float* qp = q + (size_t)b * DIM + c0;
  float qv[DH];
#pragma unroll
  for (int j = 0; j < DH; ++j) qv[j] = qp[j];
  const float* Kb = Katt + (size_t)b * NODES * DIM + c0;
  const float* Vb = Vatt + (size_t)b * NODES * DIM + c0;
  const int* mb = mask + (size_t)b * NODES;
  float m = -1e30f, l = 0.f, acc[DH];
#pragma unroll
  for (int j = 0; j < DH; ++j) acc[j] = 0.f;
  for (int i = 0; i < 32; ++i) {
    const int t = lane + (i << 5);
    if (t < NODES) {
      if (t + 32 < NODES) {
        __builtin_prefetch(Kb + (size_t)(t + 32) * DIM, 0, 0);
        __builtin_prefetch(Vb + (size_t)(t + 32) * DIM, 0, 0);
      }
      const float* kr = Kb + (size_t)t * DIM;
      const float* vr = Vb + (size_t)t * DIM;
      float kk[DH];
#pragma unroll
      for (int j = 0; j < 4; ++j) {
        float4 tv = ((const float4*)kr)[j];
        kk[4 * j] = tv.x; kk[4 * j + 1] = tv.y; kk[4 * j + 2] = tv.z; kk[4 * j + 3] = tv.w;
      }
      float s = 0.f;
#pragma unroll
      for (int j = 0; j < DH; ++j) s = fmaf(qv[j], kk[j], s);
      s *= 0.25f;
      if (mb[t] != 0) s = -1e9f;
      const float mn = fmaxf(m, s);
      const float p  = __expf(s - mn);
      const float co = __expf(m - mn);
      float vv[DH];
#pragma unroll
      for (int j = 0; j < 4; ++j) {
        float4 tv = ((const float4*)vr)[j];
        vv[4 * j] = tv.x; vv[4 * j + 1] = tv.y; vv[4 * j + 2] = tv.z; vv[4 * j + 3] = tv.w;
      }
      l = l * co + p;
#pragma unroll
      for (int j = 0; j < DH; ++j) acc[j] = fmaf(acc[j], co, p * vv[j]);
      m = mn;
    }
  }
  const float M = wred_max(m);
  const float f = __expf(m - M);
  const float L = wred_sum(l * f);
#pragma unroll
  for (int j = 0; j < DH; ++j) red[wave][lane][j] = acc[j] * f;
  __syncthreads();
  if (lane < DH) {
    float o = 0.f;
#pragma unroll 8
    for (int i = 0; i < 32; ++i) o += red[wave][i][lane];
    out[(size_t)b * DIM + c0 + lane] = o / L;
  }
}

// Row-wise LayerNorm over 128 channels; one block (128 thr) per row.
__global__ __launch_bounds__(128) void layernorm_kernel(
    const float* __restrict__ in, const float* __restrict__ g,
    const float* __restrict__ be, float* __restrict__ out) {
  __shared__ float ps[4], pq[4];
  const int row = blockIdx.x, tid = threadIdx.x;
  const int wave = tid >> 5, lane = tid & 31;
  const float x = in[(size_t)row * DIM + tid];
  float s = wred_sum(x);
  if (lane == 0) ps[wave] = s;
  __syncthreads();
  const float mu = (ps[0] + ps[1] + ps[2] + ps[3]) * (1.f / DIM);
  const float d = x - mu;
  float sq = wred_sum(d * d);
  if (lane == 0) pq[wave] = sq;
  __syncthreads();
  const float var = (pq[0] + pq[1] + pq[2] + pq[3]) * (1.f / DIM);
  out[(size_t)row * DIM + tid] = d * rsqrtf(var + 1e-5f) * g[tid] + be[tid];
}

extern "C" void kernel_launch(void* const* d_in, const int* in_sizes, int n_in,
                              void* d_out, int out_size, void* d_ws, size_t ws_size,
                              hipStream_t stream) {
  (void)in_sizes; (void)n_in; (void)out_size; (void)ws_size;
  const float* h_t   = (const float*)d_in[0];
  const float* K_att = (const float*)d_in[1];
  const float* V_att = (const float*)d_in[2];
  const int*   mask  = (const int*)d_in[3];
  const float* K_sa  = (const float*)d_in[4];
  const float* V_sa  = (const float*)d_in[5];
  const float* Wq_sa = (const float*)d_in[6];
  const float* bq_sa = (const float*)d_in[7];
  const float* Wk_sa = (const float*)d_in[8];
  const float* bk_sa = (const float*)d_in[9];
  const float* Wv_sa = (const float*)d_in[10];
  const float* bv_sa = (const float*)d_in[11];
  const float* W0_sa = (const float*)d_in[12];
  const float* b0_sa = (const float*)d_in[13];
  const float* Wq_a  = (const float*)d_in[14];
  const float* bq_a  = (const float*)d_in[15];
  const float* W0_a  = (const float*)d_in[16];
  const float* b0_a  = (const float*)d_in[17];
  const float* W1    = (const float*)d_in[18];
  const float* b1    = (const float*)d_in[19];
  const float* W2    = (const float*)d_in[20];
  const float* b2    = (const float*)d_in[21];
  const float* g_sa  = (const float*)d_in[22];
  const float* g_a   = (const float*)d_in[23];
  const float* g_m   = (const float*)d_in[24];
  const float* be_sa = (const float*)d_in[25];
  const float* be_a  = (const float*)d_in[26];
  const float* be_m  = (const float*)d_in[27];

  float* ws = (float*)d_ws;
  const size_t SLOT = (size_t)BSZ * DIM;
  float* s0 = ws;            float* s1 = ws + SLOT;
  float* s2 = ws + 2 * SLOT; float* s3 = ws + 3 * SLOT;
  float* s4 = ws + 4 * SLOT; float* s5 = ws + 5 * SLOT;

  dim3 gG(BSZ / 16), bG(256);
  dim3 gA(BSZ), bA(256);
  dim3 gL(BSZ), bL(128);

  // q/k/v projections
  gemm128_wmma<0, 0><<<gG, bG, 0, stream>>>(h_t, Wq_sa, bq_sa, nullptr, s0);
  gemm128_wmma<0, 0><<<gG, bG, 0, stream>>>(h_t, Wk_sa, bk_sa, nullptr, s1);
  gemm128_wmma<0, 0><<<gG, bG, 0, stream>>>(h_t, Wv_sa, bv_sa, nullptr, s2);
  // self-attention over 511 cached + 1 appended token
  self_attn_kernel<<<gA, bA, 0, stream>>>(s0, K_sa, V_sa, s1, s2, s3);
  // h1 = LN(h + attn @ W0_sa + b0_sa)
  gemm128_wmma<0, 1><<<gG, bG, 0, stream>>>(s3, W0_sa, b0_sa, h_t, s4);
  layernorm_kernel<<<gL, bL, 0, stream>>>(s4, g_sa, be_sa, s5);
  // cross-attention with mask
  gemm128_wmma<0, 0><<<gG, bG, 0, stream>>>(s5, Wq_a, bq_a, nullptr, s0);
  cross_attn_kernel<<<gA, bA, 0, stream>>>(s0, K_att, V_att, mask, s1);
  // h2 = LN(h1 + attn @ W0_a + b0_a)
  gemm128_wmma<0, 1><<<gG, bG, 0, stream>>>(s1, W0_a, b0_a, s5, s2);
  layernorm_kernel<<<gL, bL, 0, stream>>>(s2, g_a, be_a, s3);
  // MLP + residual + final LN
  gemm128_wmma<1, 0><<<gG, bG, 0, stream>>>(s3, W1, b1, nullptr, s4);
  gemm128_wmma<0, 1><<<gG, bG, 0, stream>>>(s4, W2, b2, s3, s0);
  layernorm_kernel<<<gL, bL, 0, stream>>>(s0, g_m, be_m, (float*)d_out);
}